// ALSHConv2d_26645977104466
// MI455X (gfx1250) — compile-verified
//
#include <hip/hip_runtime.h>
#include <math.h>

#define SPAN 2304
#define IN_C 256
#define OUT_C 512
#define HW 3136          // 56*56
#define W2 56
#define NIMG 16
#define LCOLS (NIMG*HW)  // 50176
#define TABLE_SIZE 8
#define RBIN 2.5f
#define KC 64            // K-chunk per LDS stage
#define BN 128           // columns per block (8 waves * 16)
#define MT 32            // rows per block (2 x 16-row WMMA tiles per wave)
// k-pair interleaved layouts: element (k,c) at (k>>1)*STR + 2*c + (k&1)
// STR % 64 == 32 so the two half-wave k-pairs hit disjoint 32-bank halves.
#define ASTR 96          // A row-pair stride in dwords (2*32 data + 32 pad)
#define BSTR 288         // B row-pair stride in dwords (2*128 data + 32 pad)

// ws layout (int32 slots): [0..7] bucket counts, [8] oc_eff, [9] oc (rows out count),
// [10] bucket, [12](float) scale, [16..527] rows, [528..1039] kbuck
#define WS_OC     8
#define WS_NROWS  9
#define WS_BUCKET 10
#define WS_SCALE  12
#define WS_ROWS   16
#define WS_KBUCK  528

typedef float v2f __attribute__((ext_vector_type(2)));
typedef float v8f __attribute__((ext_vector_type(8)));

__device__ __forceinline__ int lsh_bucket(float v, float b) {
    int idx = (int)floorf((v + b) / RBIN);
    int bk = idx % TABLE_SIZE;           // fmod: sign of dividend
    return bk < 0 ? -bk : bk;            // then abs
}

// ---------------- kernel 1: bucket every kernel row (one wave32 per row) ------
__global__ __launch_bounds__(256)
void kbuck_kernel(const float* __restrict__ kern,
                  const float* __restrict__ a_hash,
                  const float* __restrict__ b_hash,
                  int* __restrict__ ws_i) {
    if (blockIdx.x == 0 && threadIdx.x < TABLE_SIZE) ws_i[threadIdx.x] = 0; // zero counts each call
    int wave = (blockIdx.x * blockDim.x + threadIdx.x) >> 5;
    int lane = threadIdx.x & 31;
    if (wave >= OUT_C) return;
    const float* row = kern + (size_t)wave * SPAN;
    float dot = 0.f, nn = 0.f;
    for (int s = lane; s < SPAN; s += 32) {
        float v = row[s];
        dot = fmaf(v, a_hash[s], dot);
        nn  = fmaf(v, v, nn);
    }
    #pragma unroll
    for (int m = 16; m; m >>= 1) {
        dot += __shfl_xor(dot, m, 32);
        nn  += __shfl_xor(nn,  m, 32);
    }
    if (lane == 0) {
        float n2 = nn, n4 = n2 * n2, n8 = n4 * n4;   // ||x||^2, ^4, ^8
        float val = dot + n2 * a_hash[SPAN] + n4 * a_hash[SPAN + 1] + n8 * a_hash[SPAN + 2];
        ws_i[WS_KBUCK + wave] = lsh_bucket(val, b_hash[0]);
    }
}

// ---------------- kernel 2: hash every unit patch column, vote ---------------
__global__ __launch_bounds__(256)
void vote_kernel(const float* __restrict__ inp,
                 const float* __restrict__ a_hash,
                 const float* __restrict__ b_hash,
                 int* __restrict__ ws_i) {
    int col = blockIdx.x * blockDim.x + threadIdx.x;
    if (col >= LCOLS) return;
    int n = col / HW, p = col - n * HW;
    int py = p / W2, px = p - py * W2;
    const float* base = inp + (size_t)n * IN_C * HW;
    float dot = 0.f, nn = 0.f;
    #pragma unroll
    for (int kh = 0; kh < 3; ++kh) {
        int y = py + kh - 1;
        if ((unsigned)y >= (unsigned)W2) continue;
        #pragma unroll
        for (int kw = 0; kw < 3; ++kw) {
            int x = px + kw - 1;
            if ((unsigned)x >= (unsigned)W2) continue;
            int off = kh * 3 + kw;
            const float* q = base + y * W2 + x;
            for (int c = 0; c < IN_C; ++c) {
                float v = q[(size_t)c * HW];
                dot = fmaf(v, a_hash[c * 9 + off], dot);
                nn  = fmaf(v, v, nn);
            }
        }
    }
    float ext = 0.5f * (a_hash[SPAN] + a_hash[SPAN + 1] + a_hash[SPAN + 2]);
    float vote = dot * rsqrtf(nn) + ext;           // a . (col/||col||) + ext
    atomicAdd(&ws_i[lsh_bucket(vote, b_hash[0])], 1);
}

// ---------------- kernel 3: argmax bucket, compact active rows ---------------
__global__ void finalize_kernel(const int* __restrict__ mode,
                                int* __restrict__ ws_i,
                                float* __restrict__ ws_f,
                                float* __restrict__ d_out) {
    if (threadIdx.x != 0 || blockIdx.x != 0) return;
    int best = 0, bc = ws_i[0];
    for (int i = 1; i < TABLE_SIZE; ++i)
        if (ws_i[i] > bc) { bc = ws_i[i]; best = i; }   // first-max tie break
    int oc = 0;
    for (int r = 0; r < OUT_C; ++r)
        if (ws_i[WS_KBUCK + r] == best) ws_i[WS_ROWS + oc++] = r;  // ascending
    int oc_eff = oc;
    float scale = 1.f;
    if (oc == 0) {                       // empty bucket -> dense GEMM, no scale
        oc_eff = OUT_C;
        for (int r = 0; r < OUT_C; ++r) ws_i[WS_ROWS + r] = r;
    } else if (mode[0]) {
        scale = (float)OUT_C / (float)oc;
    }
    ws_i[WS_OC]     = oc_eff;
    ws_i[WS_NROWS]  = oc;
    ws_i[WS_BUCKET] = best;
    ws_f[WS_SCALE]  = scale;
    // rows tail of d_out (int32 bit pattern appended after N*oc_eff*HW floats)
    int* rows_out = (int*)d_out + (size_t)NIMG * oc_eff * HW;
    for (int i = 0; i < oc; ++i) rows_out[i] = ws_i[WS_ROWS + i];
}

// ---------------- kernel 4: WMMA fp32 GEMM, implicit im2col ------------------
// C[32 x 128] per block (two 16-row WMMA tiles share each B fragment).
// A/B staged in k-pair-interleaved LDS so every fragment is one ds_load_b64.
__global__ __launch_bounds__(256)
void gemm_kernel(const float* __restrict__ inp,
                 const float* __restrict__ kern,
                 const int* __restrict__ ws_i,
                 const float* __restrict__ ws_f,
                 float* __restrict__ out) {
    const int oc_eff = ws_i[WS_OC];
    const int tileM = blockIdx.y;
    if (tileM * MT >= oc_eff) return;           // dynamic M: skip dead row tiles
    const float scale = ws_f[WS_SCALE];

    __shared__ __align__(16) float Alds[(KC / 2) * ASTR];   // (k,m) -> (k>>1)*ASTR + 2m + (k&1)
    __shared__ __align__(16) float Blds[(KC / 2) * BSTR];   // (k,c) -> (k>>1)*BSTR + 2c + (k&1)
    __shared__ int rowsbuf[MT];

    const int t = threadIdx.x;
    if (t < MT) {
        int mrow = tileM * MT + t;
        rowsbuf[t] = (mrow < oc_eff) ? ws_i[WS_ROWS + mrow] : -1;
    }
    __syncthreads();

    const int lane = t & 31;
    const int wave = t >> 5;
    const int colbase = blockIdx.x * BN;
    const int k2h = lane >> 4;            // 0 (lanes 0-15) or 1 (lanes 16-31): k-pair select
    const int ml = lane & 15;
    const int n0 = wave * 16;

    v8f acc0 = {};    // rows tileM*MT + 0..15
    v8f acc1 = {};    // rows tileM*MT + 16..31

    for (int kb = 0; kb < SPAN; kb += KC) {
        // --- stage A chunk: 32 gathered rows x KC (2048 elems / 256 thr) ---
        #pragma unroll
        for (int i = 0; i < (MT * KC) / 256; ++i) {
            int e = t + i * 256;
            int k = e >> 5, m = e & 31;
            int r = rowsbuf[m];
            float v = (r >= 0) ? kern[(size_t)r * SPAN + kb + k] : 0.f;
            Alds[(k >> 1) * ASTR + (m << 1) + (k & 1)] = v;
        }
        // --- stage B chunk: KC x 128 im2col (8192 elems / 256 thr) ---
        #pragma unroll
        for (int i = 0; i < (KC * BN) / 256; ++i) {
            int e = t + i * 256;
            int kl = e >> 7, ci = e & (BN - 1);
            int col = colbase + ci;
            int n = col / HW, p = col - n * HW;
            int py = p / W2, px = p - py * W2;
            int s = kb + kl;
            int c = s / 9, rm = s - c * 9;
            int kh = rm / 3, kw = rm - kh * 3;
            int y = py + kh - 1, x = px + kw - 1;
            float v = 0.f;
            if ((unsigned)y < (unsigned)W2 && (unsigned)x < (unsigned)W2)
                v = inp[(size_t)(n * IN_C + c) * HW + y * W2 + x];
            Blds[(kl >> 1) * BSTR + (ci << 1) + (kl & 1)] = v;
        }
        __syncthreads();

        // per-lane fragment base addresses (k-pair j = kk/2 + k2h)
        const float* aP = &Alds[k2h * ASTR + (ml << 1)];
        const float* bP = &Blds[k2h * BSTR + ((n0 + ml) << 1)];

        #pragma unroll
        for (int kk = 0; kk < KC; kk += 4) {
            const int j = (kk >> 1);      // k-pair row advance (2 pairs per step)
            v2f b  = *(const v2f*)(bP + j * BSTR);
            v2f a0 = *(const v2f*)(aP + j * ASTR);
            v2f a1 = *(const v2f*)(aP + j * ASTR + 32);
            acc0 = __builtin_amdgcn_wmma_f32_16x16x4_f32(
                false, a0, false, b, (short)0, acc0, false, false);
            acc1 = __builtin_amdgcn_wmma_f32_16x16x4_f32(
                false, a1, false, b, (short)0, acc1, false, false);
        }
        __syncthreads();
    }

    // epilogue: C vgpr r -> M=r (lanes 0-15) / M=r+8 (lanes 16-31), N=ml
    int colg = colbase + n0 + ml;
    int n = colg / HW, p = colg - n * HW;
    size_t obase = (size_t)n * oc_eff * HW + p;
    int madd = (lane >> 4) << 3;
    #pragma unroll
    for (int r = 0; r < 8; ++r) {
        int mrow0 = tileM * MT + r + madd;
        int mrow1 = mrow0 + 16;
        if (mrow0 < oc_eff)
            out[obase + (size_t)mrow0 * HW] = acc0[r] * scale;
        if (mrow1 < oc_eff)
            out[obase + (size_t)mrow1 * HW] = acc1[r] * scale;
    }
}

extern "C" void kernel_launch(void* const* d_in, const int* in_sizes, int n_in,
                              void* d_out, int out_size, void* d_ws, size_t ws_size,
                              hipStream_t stream) {
    const float* inp    = (const float*)d_in[0];
    const float* kern   = (const float*)d_in[1];
    const float* a_hash = (const float*)d_in[2];
    const float* b_hash = (const float*)d_in[3];
    const int*   mode   = (const int*)d_in[4];
    int*   ws_i = (int*)d_ws;
    float* ws_f = (float*)d_ws;
    float* out  = (float*)d_out;

    kbuck_kernel<<<OUT_C / 8, 256, 0, stream>>>(kern, a_hash, b_hash, ws_i);
    vote_kernel<<<LCOLS / 256, 256, 0, stream>>>(inp, a_hash, b_hash, ws_i);
    finalize_kernel<<<1, 32, 0, stream>>>(mode, ws_i, ws_f, out);
    dim3 grid(LCOLS / BN, OUT_C / MT);
    gemm_kernel<<<grid, 256, 0, stream>>>(inp, kern, ws_i, ws_f, out);
}